// LightGCN_5626407158116
// MI455X (gfx1250) — compile-verified
//
#include <hip/hip_runtime.h>
#include <math.h>

// LightGCN inference for MI455X (gfx1250, wave32).
//
// Phases (all on `stream`, graph-capture safe):
//  1) deg[i]      = segment_sum(edge_weight over col)      -- atomicAdd
//  2) dinv[i]     = deg>0 ? rsqrt(deg) : 0
//  3) norm[e]     = dinv[row]*w*dinv[col]
//  4) w4          = softmax(layer_weights)                  (4 values)
//  5) 3x: x_{l+1}[col] += norm * x_l[row]   (wave-per-edge, f32 atomics; L2-resident)
//  6) combine+normalize: V_WMMA_F32_16X16X4_F32 contracts the 4 layers
//     (A = weights broadcast over rows, B = 4x16 data tile, D column j =
//      combined value at position j -> static per-lane extraction d[0]).

#define EMB 64

typedef __attribute__((ext_vector_type(2))) float v2f;
typedef __attribute__((ext_vector_type(8))) float v8f;

__global__ void zero_f32_kernel(float* __restrict__ p, long long n) {
  long long i = (long long)blockIdx.x * blockDim.x + threadIdx.x;
  if (i < n) p[i] = 0.0f;
}

__global__ void deg_kernel(const int* __restrict__ col, const float* __restrict__ ew,
                           float* __restrict__ deg, int E) {
  int e = blockIdx.x * blockDim.x + threadIdx.x;
  if (e < E) atomicAdd(&deg[col[e]], ew[e]);
}

__global__ void dinv_kernel(float* __restrict__ deg, int N) {
  int i = blockIdx.x * blockDim.x + threadIdx.x;
  if (i < N) {
    float d = deg[i];
    deg[i] = (d > 0.0f) ? rsqrtf(d) : 0.0f;
  }
}

__global__ void norm_kernel(const int* __restrict__ row, const int* __restrict__ col,
                            const float* __restrict__ ew, const float* __restrict__ dinv,
                            float* __restrict__ nrm, int E) {
  int e = blockIdx.x * blockDim.x + threadIdx.x;
  if (e < E) nrm[e] = dinv[row[e]] * ew[e] * dinv[col[e]];
}

__global__ void softmax4_kernel(const float* __restrict__ lw, float* __restrict__ w4) {
  if (threadIdx.x == 0 && blockIdx.x == 0) {
    float m = fmaxf(fmaxf(lw[0], lw[1]), fmaxf(lw[2], lw[3]));
    float e0 = expf(lw[0] - m), e1 = expf(lw[1] - m);
    float e2 = expf(lw[2] - m), e3 = expf(lw[3] - m);
    float inv = 1.0f / (e0 + e1 + e2 + e3);
    w4[0] = e0 * inv; w4[1] = e1 * inv; w4[2] = e2 * inv; w4[3] = e3 * inv;
  }
}

// One wave32 per edge: lane l handles feature pair (2l, 2l+1).
// Gather is a coalesced 256B row read; scatter is 2 f32 atomics/lane into L2.
__global__ __launch_bounds__(256) void scatter_layer_kernel(
    const int* __restrict__ row, const int* __restrict__ col,
    const float* __restrict__ nrm, const float* __restrict__ xin,
    float* __restrict__ xout, int E) {
  int wid  = blockIdx.x * (blockDim.x >> 5) + (threadIdx.x >> 5);
  int lane = threadIdx.x & 31;
  if (wid >= E) return;
  int r = row[wid];
  int c = col[wid];
  float s = nrm[wid];
  const float2* src = (const float2*)(xin + (long long)r * EMB);
  float2 v = src[lane];
  float* dst = xout + (long long)c * EMB + lane * 2;
  atomicAdd(dst + 0, s * v.x);
  atomicAdd(dst + 1, s * v.y);
}

// One wave32 per node. 4 WMMAs (16 cols each) contract the 4 layers.
// A(16x4): every row = softmax weights  -> lane<16: {w0,w1}, lane>=16: {w2,w3}
// B(4x16): column j = 4 layer values at position base+16t+j
// D: column j identical over rows -> lane L reads comb[base+16t+(L&15)] = d_t[0].
__global__ __launch_bounds__(256) void combine_normalize_kernel(
    const float* __restrict__ x0, const float* __restrict__ x1,
    const float* __restrict__ x2, const float* __restrict__ x3,
    const float* __restrict__ w4, float* __restrict__ out, int N) {
  int node = blockIdx.x * (blockDim.x >> 5) + (threadIdx.x >> 5);
  int lane = threadIdx.x & 31;
  if (node >= N) return;

  bool hi  = lane >= 16;
  int cpos = lane & 15;

  v2f a;
  a[0] = hi ? w4[2] : w4[0];
  a[1] = hi ? w4[3] : w4[1];
  const float* bl0 = hi ? x2 : x0;   // low lane-half carries layers 0,1
  const float* bl1 = hi ? x3 : x1;   // high lane-half carries layers 2,3

  long long base = (long long)node * EMB;

  v2f b0, b1, b2, b3;
  b0[0] = bl0[base +  0 + cpos];  b0[1] = bl1[base +  0 + cpos];
  b1[0] = bl0[base + 16 + cpos];  b1[1] = bl1[base + 16 + cpos];
  b2[0] = bl0[base + 32 + cpos];  b2[1] = bl1[base + 32 + cpos];
  b3[0] = bl0[base + 48 + cpos];  b3[1] = bl1[base + 48 + cpos];

  v8f d0 = {}, d1 = {}, d2 = {}, d3 = {};
  d0 = __builtin_amdgcn_wmma_f32_16x16x4_f32(false, a, false, b0, (short)0, d0, false, false);
  d1 = __builtin_amdgcn_wmma_f32_16x16x4_f32(false, a, false, b1, (short)0, d1, false, false);
  d2 = __builtin_amdgcn_wmma_f32_16x16x4_f32(false, a, false, b2, (short)0, d2, false, false);
  d3 = __builtin_amdgcn_wmma_f32_16x16x4_f32(false, a, false, b3, (short)0, d3, false, false);

  float c0 = d0[0], c1 = d1[0], c2 = d2[0], c3 = d3[0];

  // Row sum of squares: per-lane partial over its 4 column-class values,
  // then butterfly over the 16 columns (lane halves hold identical data).
  float ss = c0 * c0 + c1 * c1 + c2 * c2 + c3 * c3;
  ss += __shfl_xor(ss, 1, 32);
  ss += __shfl_xor(ss, 2, 32);
  ss += __shfl_xor(ss, 4, 32);
  ss += __shfl_xor(ss, 8, 32);

  float scale = 1.0f / fmaxf(sqrtf(ss), 1e-12f);

  if (!hi) {
    out[base +  0 + cpos] = c0 * scale;
    out[base + 16 + cpos] = c1 * scale;
  } else {
    out[base + 32 + cpos] = c2 * scale;
    out[base + 48 + cpos] = c3 * scale;
  }
}

extern "C" void kernel_launch(void* const* d_in, const int* in_sizes, int n_in,
                              void* d_out, int out_size, void* d_ws, size_t ws_size,
                              hipStream_t stream) {
  const float* emb = (const float*)d_in[0];   // [N, 64] f32
  const float* lw  = (const float*)d_in[1];   // [4] f32
  const float* ew  = (const float*)d_in[2];   // [E] f32
  const int*   ei  = (const int*)d_in[3];     // [2, E] int32

  int N = in_sizes[0] / EMB;
  int E = in_sizes[2];
  const int* row = ei;       // source
  const int* col = ei + E;   // target

  // Workspace carving (256B aligned).
  char* ws = (char*)d_ws;
  size_t off = 0;
  auto carve = [&](size_t bytes) -> void* {
    void* p = ws + off;
    off = (off + bytes + 255) & ~(size_t)255;
    return p;
  };
  float* deg  = (float*)carve((size_t)N * sizeof(float));          // deg -> dinv in place
  float* nrm  = (float*)carve((size_t)E * sizeof(float));
  float* w4   = (float*)carve(256);
  float* x1   = (float*)carve((size_t)N * EMB * sizeof(float));
  float* x2   = (float*)carve((size_t)N * EMB * sizeof(float));
  float* x3   = (float*)carve((size_t)N * EMB * sizeof(float));
  float* xbuf[3] = {x1, x2, x3};

  const int T = 256;
  long long ND = (long long)N * EMB;

  // 1) degrees
  zero_f32_kernel<<<(int)((N + T - 1) / T), T, 0, stream>>>(deg, N);
  deg_kernel<<<(E + T - 1) / T, T, 0, stream>>>(col, ew, deg, E);
  // 2) dinv
  dinv_kernel<<<(N + T - 1) / T, T, 0, stream>>>(deg, N);
  // 3) per-edge norm
  norm_kernel<<<(E + T - 1) / T, T, 0, stream>>>(row, col, ew, deg, nrm, E);
  // 4) softmax of layer weights
  softmax4_kernel<<<1, 32, 0, stream>>>(lw, w4);

  // 5) 3 propagation layers (wave-per-edge)
  const float* xin = emb;
  int wavesPerBlock = T >> 5;
  int scatterBlocks = (E + wavesPerBlock - 1) / wavesPerBlock;
  for (int l = 0; l < 3; ++l) {
    float* xout = xbuf[l];
    zero_f32_kernel<<<(int)((ND + T - 1) / T), T, 0, stream>>>(xout, ND);
    scatter_layer_kernel<<<scatterBlocks, T, 0, stream>>>(row, col, nrm, xin, xout, E);
    xin = xout;
  }

  // 6) WMMA layer-combine + L2 row normalize (wave-per-node)
  int combineBlocks = (N + wavesPerBlock - 1) / wavesPerBlock;
  combine_normalize_kernel<<<combineBlocks, T, 0, stream>>>(
      emb, x1, x2, x3, w4, (float*)d_out, N);
}